// LieGen_33973191312031
// MI455X (gfx1250) — compile-verified
//
#include <hip/hip_runtime.h>

// LieGen elementwise kernel: per-row 2x2 rotation (angle = delta[i]*a) + lag update.
// Memory-bound: 32 B/row, ~134 MB total, floor ~5.8 us @ 23.3 TB/s HBM.
// Working set (128 MB) fits in the 192 MB L2 -> default RT cache hints.
// All addressing is 32-bit per-lane offsets over SGPR bases (GVS mode) to
// minimize VALU address math; all VMEM is b128, fully coalesced.

#define THREADS 256
#define ROWS_PER_THREAD 4

__global__ __launch_bounds__(THREADS) void liegen_rot_kernel(
    const float* __restrict__ z_s,    // (B,2) interleaved
    const float* __restrict__ delta,  // (B,)
    const float* __restrict__ z_l,    // (B,)
    const float* __restrict__ dt,     // (B,)
    const float* __restrict__ W,      // (2,2) row-major: W[1,0]=W[2], W[0,1]=W[1]
    const float* __restrict__ a_lag,  // (1,)
    float*       __restrict__ out,    // (B,3) interleaved
    unsigned B)
{
    // Uniform scalars (SALU path): a = A[1,0] = 0.5*(W[1,0]-W[0,1])
    const float a  = 0.5f * (W[2] - W[1]);
    const float al = a_lag[0];

    const unsigned t    = blockIdx.x * THREADS + threadIdx.x;   // vec4 index
    const unsigned row0 = t * ROWS_PER_THREAD;
    if (row0 >= B) return;

    if (row0 + ROWS_PER_THREAD <= B) {
        // 32-bit indices over SGPR bases -> GVS global addressing, b128 VMEM.
        const float4* zs_v  = (const float4*)z_s;    // 2 vec4 per thread
        const float4* d_v   = (const float4*)delta;
        const float4* zl_v  = (const float4*)z_l;
        const float4* dt_v  = (const float4*)dt;
        float4*       out_v = (float4*)out;          // 3 vec4 per thread

        const float4 zs01 = zs_v[2u * t];            // rows 0,1: z0,z1,z0,z1
        const float4 zs23 = zs_v[2u * t + 1u];       // rows 2,3
        const float4 d4   = d_v[t];
        const float4 zl4  = zl_v[t];
        const float4 dt4  = dt_v[t];

        const float z0[4] = { zs01.x, zs01.z, zs23.x, zs23.z };
        const float z1[4] = { zs01.y, zs01.w, zs23.y, zs23.w };
        const float dd[4] = { d4.x,  d4.y,  d4.z,  d4.w  };
        const float zl[4] = { zl4.x, zl4.y, zl4.z, zl4.w };
        const float tt[4] = { dt4.x, dt4.y, dt4.z, dt4.w };

        float o[12];
        #pragma unroll
        for (int k = 0; k < 4; ++k) {
            const float theta = dd[k] * a;
            const float c = __cosf(theta);   // v_cos_f32 (TRANS, co-executes with VALU)
            const float s = __sinf(theta);   // v_sin_f32
            o[k * 3 + 0] = fmaf(c, z0[k], -s * z1[k]);
            o[k * 3 + 1] = fmaf(s, z0[k],  c * z1[k]);
            o[k * 3 + 2] = fmaf(tt[k], al, zl[k]);
        }

        // 12 contiguous floats at byte offset 48*t (16B-aligned) -> 3x b128 stores.
        out_v[3u * t + 0u] = make_float4(o[0], o[1], o[2],  o[3]);
        out_v[3u * t + 1u] = make_float4(o[4], o[5], o[6],  o[7]);
        out_v[3u * t + 2u] = make_float4(o[8], o[9], o[10], o[11]);
    } else {
        // Scalar tail (never hit for B = 4M; kept for generality).
        for (unsigned r = row0; r < B; ++r) {
            const float theta = delta[r] * a;
            const float c = __cosf(theta);
            const float s = __sinf(theta);
            const float y0 = z_s[2u * r + 0u];
            const float y1 = z_s[2u * r + 1u];
            out[3u * r + 0u] = fmaf(c, y0, -s * y1);
            out[3u * r + 1u] = fmaf(s, y0,  c * y1);
            out[3u * r + 2u] = fmaf(dt[r], al, z_l[r]);
        }
    }
}

extern "C" void kernel_launch(void* const* d_in, const int* in_sizes, int n_in,
                              void* d_out, int out_size, void* d_ws, size_t ws_size,
                              hipStream_t stream) {
    const float* z_s   = (const float*)d_in[0];  // (B,2)
    const float* delta = (const float*)d_in[1];  // (B,)
    const float* z_l   = (const float*)d_in[2];  // (B,1)
    const float* dt    = (const float*)d_in[3];  // (B,)
    const float* W     = (const float*)d_in[4];  // (2,2)
    const float* a_lag = (const float*)d_in[5];  // (1,)
    float* out = (float*)d_out;                  // (B,3)

    const unsigned B = (unsigned)in_sizes[1];    // delta has B elements
    const unsigned rows_per_block = THREADS * ROWS_PER_THREAD;
    const unsigned blocks = (B + rows_per_block - 1u) / rows_per_block;

    liegen_rot_kernel<<<blocks, THREADS, 0, stream>>>(z_s, delta, z_l, dt, W, a_lag, out, B);
}